// LSHSelfAttention_69672959475822
// MI455X (gfx1250) — compile-verified
//
#include <hip/hip_runtime.h>
#include <hip/hip_bf16.h>
#include <stdint.h>

typedef __attribute__((ext_vector_type(2))) float v2f;
typedef __attribute__((ext_vector_type(8))) float v8f;

#define WMMA_F32X4(a, b, c) \
  __builtin_amdgcn_wmma_f32_16x16x4_f32(false, (a), false, (b), (short)0, (c), false, false)

// --- gfx1250 async global->LDS copy (ASYNCcnt-tracked) ----------------------
static __device__ __forceinline__ uint32_t lds_off(const void* p) {
  // Flat LDS addresses: low 32 bits are the wave-relative LDS byte address.
  return (uint32_t)(uintptr_t)p;
}
static __device__ __forceinline__ void async_ld_b128(uint32_t lds, uint64_t gaddr) {
  asm volatile("global_load_async_to_lds_b128 %0, %1, off" :: "v"(lds), "v"(gaddr) : "memory");
}
static __device__ __forceinline__ void wait_async0() {
#if __has_builtin(__builtin_amdgcn_s_wait_asynccnt)
  __builtin_amdgcn_s_wait_asynccnt(0);
#else
  asm volatile("s_wait_asynccnt 0x0" ::: "memory");
#endif
}

// ---------------------------------------------------------------------------
// NT GEMM: C[M,N] = A[M,K] * B[N,K]^T (+bias), fp32 exact via
// V_WMMA_F32_16X16X4_F32.  Block: 256 threads = 8 waves; block tile 128x128;
// wave tile 32x64.  K staged through LDS in 32-wide double-buffered blocks,
// filled by async global->LDS b128 transfers overlapped with WMMA compute.
// LDS rows padded to 36 floats (bank-conflict-free b64 fragment reads,
// 16B-aligned stores).
// ---------------------------------------------------------------------------
#define LDT 36
__global__ __launch_bounds__(256) void gemm_nt_f32(const float* __restrict__ A,
                                                   const float* __restrict__ Bm,
                                                   const float* __restrict__ bias,
                                                   float* __restrict__ C,
                                                   int M, int N, int K) {
  __shared__ float As[2][128 * LDT];
  __shared__ float Bs[2][128 * LDT];
  const int tid = threadIdx.x;
  const int lane = tid & 31;
  const int wave = tid >> 5;
  const int mw = wave & 3;   // 0..3 (M strips of 32)
  const int nh = wave >> 2;  // 0..1 (N halves of 64)
  const int mBlock = blockIdx.y * 128;
  const int nBlock = blockIdx.x * 128;

  const int r = lane & 15;
  const int ko = (lane < 16) ? 0 : 2;
  const int rowAdd = (lane < 16) ? 0 : 8;
  const int m0 = mw * 32;
  const int n0 = nh * 64;

  // Each thread moves 4 b128 for A and 4 for B per K-block (128 rows x 8 float4).
  const int srow = tid >> 1;              // 0..127
  const int sc4a = (tid & 1) * 4;         // float4 slots {0..3} or {4..7}
#define ISSUE_ASYNC(buf_, kb_)                                                        \
  {                                                                                   \
    _Pragma("unroll") for (int i = 0; i < 4; ++i) {                                   \
      int c4 = sc4a + i;                                                              \
      async_ld_b128(lds_off(&As[buf_][srow * LDT + c4 * 4]),                          \
                    (uint64_t)(uintptr_t)(A + (size_t)(mBlock + srow) * K + (kb_) + c4 * 4)); \
      async_ld_b128(lds_off(&Bs[buf_][srow * LDT + c4 * 4]),                          \
                    (uint64_t)(uintptr_t)(Bm + (size_t)(nBlock + srow) * K + (kb_) + c4 * 4)); \
    }                                                                                 \
  }

  ISSUE_ASYNC(0, 0);
  v8f acc[2][4] = {};
  wait_async0();
  __syncthreads();

  int buf = 0;
  for (int kb = 0; kb < K; kb += 32) {
    const int nxt = kb + 32;
    if (nxt < K) ISSUE_ASYNC(buf ^ 1, nxt);  // stream next tiles while computing

    const float* as = As[buf];
    const float* bs = Bs[buf];
#pragma unroll
    for (int k = 0; k < 32; k += 4) {
      v2f a0 = *(const v2f*)&as[(m0 + r) * LDT + k + ko];
      v2f a1 = *(const v2f*)&as[(m0 + 16 + r) * LDT + k + ko];
#pragma unroll
      for (int j = 0; j < 4; ++j) {
        v2f bf = *(const v2f*)&bs[(n0 + j * 16 + r) * LDT + k + ko];
        acc[0][j] = WMMA_F32X4(a0, bf, acc[0][j]);
        acc[1][j] = WMMA_F32X4(a1, bf, acc[1][j]);
      }
    }
    wait_async0();
    __syncthreads();
    buf ^= 1;
  }

#pragma unroll
  for (int mi = 0; mi < 2; ++mi) {
#pragma unroll
    for (int j = 0; j < 4; ++j) {
      int col = nBlock + n0 + j * 16 + r;
      float bb = bias ? bias[col] : 0.0f;
#pragma unroll
      for (int t = 0; t < 8; ++t) {
        C[(size_t)(mBlock + m0 + mi * 16 + t + rowAdd) * N + col] = acc[mi][j][t] + bb;
      }
    }
  }
}
#undef ISSUE_ASYNC

// ---------------------------------------------------------------------------
// LSH hashing: rotated = qk_merged . rotations, bucket = argmax([r, -r]).
// Block handles (bh, 32 tokens); thread = (token, hash) with 32 accumulators.
// rotations layout [128][8][32] row-major.
// ---------------------------------------------------------------------------
__global__ __launch_bounds__(256) void lsh_hash(const float* __restrict__ qk,
                                                const float* __restrict__ rot,
                                                int* __restrict__ buckets) {
  __shared__ float qs[32 * 128];
  const int bh = blockIdx.y;
  const int t0 = blockIdx.x * 32;
  const int b = bh >> 3, hh = bh & 7;
#pragma unroll
  for (int it = 0; it < 16; ++it) {
    int idx = it * 256 + threadIdx.x;  // 0..4095
    int tk = idx >> 7, d = idx & 127;
    qs[idx] = qk[((size_t)(b * 4096 + t0 + tk)) * 1024 + hh * 128 + d];
  }
  __syncthreads();

  const int tok = threadIdx.x >> 3;
  const int h = threadIdx.x & 7;
  float acc[32];
#pragma unroll
  for (int i = 0; i < 32; ++i) acc[i] = 0.0f;

  for (int d = 0; d < 128; ++d) {
    float q = qs[tok * 128 + d];
    const float4* bp = (const float4*)(rot + (size_t)d * 256 + h * 32);
#pragma unroll
    for (int ii = 0; ii < 8; ++ii) {
      float4 rv = bp[ii];
      acc[ii * 4 + 0] += q * rv.x;
      acc[ii * 4 + 1] += q * rv.y;
      acc[ii * 4 + 2] += q * rv.z;
      acc[ii * 4 + 3] += q * rv.w;
    }
  }
  // argmax over concat([r, -r]) with first-index tie-break.
  float vmax = -3.0e38f, vmin = 3.0e38f;
  int imax = 0, imin = 0;
#pragma unroll
  for (int i = 0; i < 32; ++i) {
    if (acc[i] > vmax) { vmax = acc[i]; imax = i; }
    if (acc[i] < vmin) { vmin = acc[i]; imin = i; }
  }
  int bkt = (vmax >= -vmin) ? imax : (32 + imin);
  buckets[((size_t)bh * 8 + h) * 4096 + (t0 + tok)] = bkt;
}

// ---------------------------------------------------------------------------
// Counting sort per (bh, hash): keys (bucket, t) are unique, hash rows are
// independent (bucket ids carry hash*64 offset in the reference).
// Emits st[bh][slot] = token and pos_of[bh][h*4096+t] = slot (undo_sort).
// ---------------------------------------------------------------------------
__global__ __launch_bounds__(256) void lsh_sort(const int* __restrict__ buckets,
                                                int* __restrict__ st,
                                                int* __restrict__ pos_of) {
  __shared__ int bk[4096];
  __shared__ int cnt[64];
  __shared__ int off[64];
  const int h = blockIdx.x, bh = blockIdx.y;
  const int* brow = buckets + ((size_t)bh * 8 + h) * 4096;
#pragma unroll
  for (int it = 0; it < 16; ++it) bk[it * 256 + threadIdx.x] = brow[it * 256 + threadIdx.x];
  if (threadIdx.x < 64) cnt[threadIdx.x] = 0;
  __syncthreads();
#pragma unroll
  for (int it = 0; it < 16; ++it) atomicAdd(&cnt[bk[it * 256 + threadIdx.x]], 1);
  __syncthreads();
  if (threadIdx.x == 0) {
    int run = 0;
    for (int j = 0; j < 64; ++j) { off[j] = run; run += cnt[j]; }
  }
  __syncthreads();
  if (threadIdx.x < 64) {
    const int j = threadIdx.x;
    int slot = h * 4096 + off[j];
    int* stp = st + (size_t)bh * 32768;
    int* pp = pos_of + (size_t)bh * 32768 + h * 4096;
    for (int t = 0; t < 4096; ++t) {
      if (bk[t] == j) { stp[slot] = t; pp[t] = slot; ++slot; }
    }
  }
}

// ---------------------------------------------------------------------------
// Chunked LSH attention for one (bh, chunk): gather K (chunk + look-back) to
// LDS, WMMA dots with post-GEMM column scaling by 1/||k|| * D^-0.5, self-mask,
// softmax (lse saved), reload V into same LDS, WMMA probs.V, 1/sum folded into
// the store.  LDS ~101 KB -> ~3 workgroups per 320 KB WGP.
// ---------------------------------------------------------------------------
#define LDK 132
__global__ __launch_bounds__(256) void lsh_attn(const float* __restrict__ qk,
                                                const float* __restrict__ vg,
                                                const int* __restrict__ st,
                                                float* __restrict__ so,
                                                float* __restrict__ slog) {
  __shared__ float KV[128 * LDK];
  __shared__ float Pt[64 * LDK];
  __shared__ float inv[128];
  __shared__ float rs[64];
  __shared__ int tok[128];

  const int c = blockIdx.x, bh = blockIdx.y;
  const int b = bh >> 3, hh = bh & 7;
  const int cprev = (c + 511) & 511;

  if (threadIdx.x < 128) {
    int j = threadIdx.x;
    int src = (j < 64) ? (c * 64 + j) : (cprev * 64 + (j - 64));
    tok[j] = st[(size_t)bh * 32768 + src];
  }
  __syncthreads();

  // gather K rows (raw; rows 0..63 double as Q)
#pragma unroll 4
  for (int it = 0; it < 64; ++it) {
    int idx = it * 256 + threadIdx.x;
    int row = idx >> 7, col = idx & 127;
    KV[row * LDK + col] = qk[((size_t)(b * 4096 + tok[row])) * 1024 + hh * 128 + col];
  }
  __syncthreads();

  if (threadIdx.x < 128) {
    const float* kr = &KV[threadIdx.x * LDK];
    float s = 0.0f;
    for (int d = 0; d < 128; ++d) s += kr[d] * kr[d];
    inv[threadIdx.x] = 1.0f / fmaxf(sqrtf(s), 1e-12f);
  }
  __syncthreads();

  const int lane = threadIdx.x & 31;
  const int wave = threadIdx.x >> 5;
  const int mw = wave >> 1;  // 0..3
  const int nq = wave & 1;   // 0..1
  const int m0 = mw * 16;
  const int r = lane & 15;
  const int ko = (lane < 16) ? 0 : 2;
  const int rowAdd = (lane < 16) ? 0 : 8;

  // dots = Q . K^T   (NT in LDS)
  v8f acc[4] = {};
#pragma unroll 4
  for (int k = 0; k < 128; k += 4) {
    v2f a = *(const v2f*)&KV[(m0 + r) * LDK + k + ko];
#pragma unroll
    for (int j = 0; j < 4; ++j) {
      v2f bf = *(const v2f*)&KV[(nq * 64 + j * 16 + r) * LDK + k + ko];
      acc[j] = WMMA_F32X4(a, bf, acc[j]);
    }
  }
  const float scale = 0.08838834764831845f;  // 128^-0.5
#pragma unroll
  for (int j = 0; j < 4; ++j) {
    int col = nq * 64 + j * 16 + r;
    float cinv = inv[col] * scale;
    int tkc = tok[col];
#pragma unroll
    for (int t = 0; t < 8; ++t) {
      int row = m0 + t + rowAdd;
      float d = acc[j][t] * cinv;
      if (tok[row] == tkc) d = -5.0e4f;
      Pt[row * LDK + col] = d;
    }
  }
  __syncthreads();

  if (threadIdx.x < 64) {
    float* pr = &Pt[threadIdx.x * LDK];
    float m = -3.0e38f;
    for (int j = 0; j < 128; ++j) m = fmaxf(m, pr[j]);
    float s = 0.0f;
    for (int j = 0; j < 128; ++j) {
      float e = expf(pr[j] - m);
      pr[j] = e;
      s += e;
    }
    rs[threadIdx.x] = 1.0f / s;
    slog[(size_t)bh * 32768 + c * 64 + threadIdx.x] = m + logf(s);
  }
  __syncthreads();

  // reload V into the same LDS buffer
#pragma unroll 4
  for (int it = 0; it < 64; ++it) {
    int idx = it * 256 + threadIdx.x;
    int row = idx >> 7, col = idx & 127;
    KV[row * LDK + col] = vg[((size_t)(b * 4096 + tok[row])) * 1024 + hh * 128 + col];
  }
  __syncthreads();

  // bo = probs . V   (NN in LDS)
  v8f acc2[4] = {};
#pragma unroll 2
  for (int k = 0; k < 128; k += 4) {
    v2f a = *(const v2f*)&Pt[(m0 + r) * LDK + k + ko];
#pragma unroll
    for (int j = 0; j < 4; ++j) {
      int col0 = nq * 64 + j * 16;
      v2f bf;
      bf.x = KV[(k + ko) * LDK + col0 + r];
      bf.y = KV[(k + ko + 1) * LDK + col0 + r];
      acc2[j] = WMMA_F32X4(a, bf, acc2[j]);
    }
  }
#pragma unroll
  for (int j = 0; j < 4; ++j) {
    int col = nq * 64 + j * 16 + r;
#pragma unroll
    for (int t = 0; t < 8; ++t) {
      int row = m0 + t + rowAdd;
      so[((size_t)bh * 32768 + c * 64 + row) * 128 + col] = acc2[j][t] * rs[row];
    }
  }
}

// ---------------------------------------------------------------------------
// Combine over hashes: w = softmax over 8 hash logits; attn_merged[b][t][h*128+d].
// Block = (bh, 8 tokens); thread = (token, 4 contiguous d via float4).
// ---------------------------------------------------------------------------
__global__ __launch_bounds__(256) void lsh_combine(const float* __restrict__ so,
                                                   const float* __restrict__ slog,
                                                   const int* __restrict__ pos_of,
                                                   float* __restrict__ attn) {
  const int bh = blockIdx.y;
  const int b = bh >> 3, hh = bh & 7;
  const int tt = threadIdx.x >> 5;
  const int lane = threadIdx.x & 31;
  const int t = blockIdx.x * 8 + tt;

  int pos[8];
  float l[8];
  float m = -3.0e38f;
#pragma unroll
  for (int h = 0; h < 8; ++h) {
    pos[h] = pos_of[(size_t)bh * 32768 + h * 4096 + t];
    l[h] = slog[(size_t)bh * 32768 + pos[h]];
    m = fmaxf(m, l[h]);
  }
  float s = 0.0f;
#pragma unroll
  for (int h = 0; h < 8; ++h) {
    l[h] = expf(l[h] - m);
    s += l[h];
  }
  const float rinv = 1.0f / s;
  float4 o = {0.0f, 0.0f, 0.0f, 0.0f};
#pragma unroll
  for (int h = 0; h < 8; ++h) {
    const float4* sp = (const float4*)(so + ((size_t)bh * 32768 + pos[h]) * 128) + lane;
    float4 x = *sp;
    float w = l[h] * rinv;
    o.x += w * x.x;
    o.y += w * x.y;
    o.z += w * x.z;
    o.w += w * x.w;
  }
  ((float4*)(attn + ((size_t)(b * 4096 + t)) * 1024 + hh * 128))[lane] = o;
}

// ---------------------------------------------------------------------------
extern "C" void kernel_launch(void* const* d_in, const int* in_sizes, int n_in,
                              void* d_out, int out_size, void* d_ws, size_t ws_size,
                              hipStream_t stream) {
  const float* x = (const float*)d_in[0];       // [4,4096,1024]
  const float* w_qk = (const float*)d_in[1];    // [1024,1024]
  const float* w_v = (const float*)d_in[2];     // [1024,1024]
  const float* w_out = (const float*)d_in[3];   // [1024,1024]
  const float* b_out = (const float*)d_in[4];   // [1024]
  const float* rot = (const float*)d_in[5];     // [128,8,32]
  float* out = (float*)d_out;                   // [4,4096,1024]

  char* ws = (char*)d_ws;
  const size_t BT_E = (size_t)16384 * 1024;          // 16.7M floats
  float* qk = (float*)ws;           ws += BT_E * 4;  // 64 MB
  float* vv = (float*)ws;           ws += BT_E * 4;  // 64 MB
  float* attn = (float*)ws;         ws += BT_E * 4;  // 64 MB
  float* so = (float*)ws;           ws += (size_t)32 * 32768 * 128 * 4;  // 512 MB
  float* slog = (float*)ws;         ws += (size_t)32 * 32768 * 4;        // 4 MB
  int* buckets = (int*)ws;          ws += (size_t)32 * 32768 * 4;        // 4 MB
  int* st = (int*)ws;               ws += (size_t)32 * 32768 * 4;        // 4 MB
  int* pos = (int*)ws;              ws += (size_t)32 * 32768 * 4;        // 4 MB

  // 1-2. qk/v projections (fp32-exact WMMA NT GEMM, async-LDS pipelined)
  gemm_nt_f32<<<dim3(8, 128), 256, 0, stream>>>(x, w_qk, nullptr, qk, 16384, 1024, 1024);
  gemm_nt_f32<<<dim3(8, 128), 256, 0, stream>>>(x, w_v, nullptr, vv, 16384, 1024, 1024);
  // 3. bucket assignment
  lsh_hash<<<dim3(128, 32), 256, 0, stream>>>(qk, rot, buckets);
  // 4. per-(bh,hash) counting sort + inverse permutation
  lsh_sort<<<dim3(8, 32), 256, 0, stream>>>(buckets, st, pos);
  // 5. chunked attention with look-back
  lsh_attn<<<dim3(512, 32), 256, 0, stream>>>(qk, vv, st, so, slog);
  // 6. combine across hash rounds into merged-head layout
  lsh_combine<<<dim3(512, 32), 256, 0, stream>>>(so, slog, pos, attn);
  // 7. output projection + bias
  gemm_nt_f32<<<dim3(8, 128), 256, 0, stream>>>(attn, w_out, b_out, out, 16384, 1024, 1024);
}